// _LibXLSTM_6640019439854
// MI455X (gfx1250) — compile-verified
//
#include <hip/hip_runtime.h>
#include <hip/hip_bf16.h>

// ---------------------------------------------------------------------------
// xLSTM-ish stack on gfx1250 (MI455X). All GEMMs via V_WMMA_F32_16X16X4_F32
// (f32 in / f32 acc) for bit-faithful-ish f32 math; the sequential sLSTM scan
// runs as persistent per-(batch-tile, head) waves with state in registers and
// weights in LDS.
// ---------------------------------------------------------------------------

typedef float v2f __attribute__((ext_vector_type(2)));
typedef float v8f __attribute__((ext_vector_type(8)));

#define XB   512
#define XS   64
#define XD   128
#define XNH  4
#define XDH  32
#define XT   (XB * XS)      // tokens
#define XEPS 1e-5f

__device__ __forceinline__ v8f wmma4(v2f a, v2f b, v8f c) {
  // 8 args: (neg_a, A, neg_b, B, c_mod, C, reuse_a, reuse_b)
  return __builtin_amdgcn_wmma_f32_16x16x4_f32(false, a, false, b, (short)0, c,
                                               false, false);
}

__device__ __forceinline__ float gelu_exact(float x) {
  return 0.5f * x * (1.0f + erff(x * 0.70710678118654752f));
}

// --------------------------- embed ----------------------------------------
__global__ void k_embed(const int* __restrict__ x, const float* __restrict__ emb,
                        float* __restrict__ h, int total) {
  int i = blockIdx.x * blockDim.x + threadIdx.x;
  if (i >= total) return;
  int t = i >> 7;
  int c = i & 127;
  h[i] = emb[x[t] * XD + c];
}

// --------------------------- layernorm (one wave per token) ---------------
__global__ void k_ln(const float* __restrict__ h, const float* __restrict__ w,
                     float* __restrict__ xn) {
  int wid  = threadIdx.x >> 5;
  int lane = threadIdx.x & 31;
  int t = blockIdx.x * (blockDim.x >> 5) + wid;
  const float4* hv = (const float4*)(h + (size_t)t * XD);
  float4 v = hv[lane];
  float s = v.x + v.y + v.z + v.w;
  float q = v.x * v.x + v.y * v.y + v.z * v.z + v.w * v.w;
  #pragma unroll
  for (int m = 16; m >= 1; m >>= 1) {
    s += __shfl_xor(s, m, 32);
    q += __shfl_xor(q, m, 32);
  }
  float mu  = s * (1.0f / XD);
  float var = q * (1.0f / XD) - mu * mu;
  float rs  = rsqrtf(var + XEPS);
  float4 wl = ((const float4*)w)[lane];
  float4 o;
  o.x = (v.x - mu) * rs * wl.x;
  o.y = (v.y - mu) * rs * wl.y;
  o.z = (v.z - mu) * rs * wl.z;
  o.w = (v.w - mu) * rs * wl.w;
  ((float4*)(xn + (size_t)t * XD))[lane] = o;
}

// --------------------------- sLSTM scan ------------------------------------
// grid = (B/16, NH), block = 32 (one wave). Each wave owns a 16-row batch tile
// for one head across all 64 timesteps. Per step: raw = bias + xn@Wg^T + h@R
// via 128 v_wmma_f32_16x16x4_f32, then the elementwise state update, then
// head groupnorm + residual into h.
__global__ __launch_bounds__(32) void k_scan(const float* __restrict__ xn,
                                             float* __restrict__ hio,
                                             const float* __restrict__ Wg,
                                             const float* __restrict__ R,
                                             const float* __restrict__ bias,
                                             const float* __restrict__ gnw) {
  __shared__ float lwg[8][8][64];   // pre-swizzled B-frags of Wg (16 KB)
  __shared__ float lr[8][8][64];    // pre-swizzled B-frags of R  (16 KB)
  __shared__ float lh[16 * 32];     // h-state staging tile       (2 KB)
  __shared__ float lb[128];         // bias (gate-major)
  __shared__ float lgn[32];         // groupnorm weight slice

  const int lane = threadIdx.x;
  const int hlf  = lane >> 4;   // 0 or 1
  const int l16  = lane & 15;
  const int b0   = blockIdx.x * 16;
  const int n    = blockIdx.y;

  // ---- prologue: swizzle weights into LDS B-fragment layout ----
  #pragma unroll
  for (int nt = 0; nt < 8; ++nt) {
    int N  = nt * 16 + l16;          // raw column j = g*32 + dh
    int g  = N >> 5;
    int dh = N & 31;
    #pragma unroll
    for (int kk = 0; kk < 8; ++kk) {
      int K0 = kk * 4 + hlf * 2;     // contraction index pair
      const float* wgp = Wg + (((size_t)(g * XNH + n) * XDH + dh) * XDH + K0);
      lwg[nt][kk][lane * 2 + 0] = wgp[0];
      lwg[nt][kk][lane * 2 + 1] = wgp[1];
      lr[nt][kk][lane * 2 + 0] = R[(size_t)(n * XDH + K0 + 0) * (4 * XDH) + N];
      lr[nt][kk][lane * 2 + 1] = R[(size_t)(n * XDH + K0 + 1) * (4 * XDH) + N];
    }
  }
  for (int j = lane; j < 128; j += 32) lb[j] = bias[n * 128 + j];
  lgn[lane & 31] = gnw[n * XDH + (lane & 31)];
  for (int j = lane; j < 16 * 32; j += 32) lh[j] = 0.0f;
  __syncthreads();

  float cS[2][8]  = {};
  float nS[2][8]  = {};
  float mS[2][8]  = {};

  #pragma unroll 1
  for (int s = 0; s < XS; ++s) {
    // A-fragments: xn rows (this step's tokens) and recurrent h from LDS
    v2f ax[8], ah[8];
    const float* xp = xn + ((size_t)(b0 + l16) * XS + s) * XD + n * XDH;
    #pragma unroll
    for (int kk = 0; kk < 8; ++kk) {
      int K0 = kk * 4 + hlf * 2;
      ax[kk].x = xp[K0];
      ax[kk].y = xp[K0 + 1];
      ah[kk].x = lh[l16 * 32 + K0];
      ah[kk].y = lh[l16 * 32 + K0 + 1];
    }

    // raw = bias + gx + ry  (8 N-tiles x 8 K-steps x 2 GEMMs)
    v8f acc[8];
    #pragma unroll
    for (int nt = 0; nt < 8; ++nt) {
      float bj = lb[nt * 16 + l16];
      v8f a;
      #pragma unroll
      for (int r = 0; r < 8; ++r) a[r] = bj;
      #pragma unroll
      for (int kk = 0; kk < 8; ++kk) {
        v2f bw, br;
        bw.x = lwg[nt][kk][lane * 2 + 0];
        bw.y = lwg[nt][kk][lane * 2 + 1];
        a = wmma4(ax[kk], bw, a);
        br.x = lr[nt][kk][lane * 2 + 0];
        br.y = lr[nt][kk][lane * 2 + 1];
        a = wmma4(ah[kk], br, a);
      }
      acc[nt] = a;
    }
    __syncthreads();  // done reading lh for this step

    // elementwise sLSTM state update; write h_new into LDS staging tile
    #pragma unroll
    for (int t2 = 0; t2 < 2; ++t2) {
      #pragma unroll
      for (int r = 0; r < 8; ++r) {
        float ir  = acc[0 + t2][r];
        float fr  = acc[2 + t2][r];
        float zr  = acc[4 + t2][r];
        float orr = acc[6 + t2][r];
        // log_sigmoid, numerically stable
        float ls  = (fr >= 0.0f) ? (-log1pf(expf(-fr)))
                                 : (fr - log1pf(expf(fr)));
        float lfm = mS[t2][r] + ls;
        float mn  = fmaxf(ir, lfm);
        float ig  = expf(ir - mn);
        float fg  = expf(lfm - mn);
        float cn  = fg * cS[t2][r] + ig * tanhf(zr);
        float nn  = fg * nS[t2][r] + ig;
        float og  = 1.0f / (1.0f + expf(-orr));
        float hn  = og * cn / nn;
        cS[t2][r] = cn;
        nS[t2][r] = nn;
        mS[t2][r] = mn;
        lh[(r + hlf * 8) * 32 + t2 * 16 + l16] = hn;
      }
    }
    __syncthreads();

    // head groupnorm (over DH=32) + residual into h
    if (lane < 16) {
      float mu = 0.0f, q = 0.0f;
      #pragma unroll
      for (int i = 0; i < 32; ++i) {
        float v = lh[lane * 32 + i];
        mu += v;
        q  += v * v;
      }
      mu *= (1.0f / XDH);
      q = q * (1.0f / XDH) - mu * mu;
      float rs = rsqrtf(q + XEPS);
      float* hp = hio + ((size_t)(b0 + lane) * XS + s) * XD + n * XDH;
      #pragma unroll
      for (int i = 0; i < 32; ++i) {
        float v = (lh[lane * 32 + i] - mu) * rs * lgn[i];
        hp[i] = hp[i] + v;
      }
    }
    __syncthreads();
  }
}

// --------------------------- gated FFN -------------------------------------
// grid = T/16, block = 32. up (16x256), act = gelu(gate)*upv, down (16x128),
// residual into h. Weights read straight from global (L2-resident, N-major so
// per-lane B-frag pairs are contiguous float2).
__global__ __launch_bounds__(32) void k_ffn(const float* __restrict__ xn,
                                            float* __restrict__ hio,
                                            const float* __restrict__ Wup,
                                            const float* __restrict__ Wdn) {
  __shared__ float lact[16 * 128];   // 8 KB
  const int lane = threadIdx.x;
  const int hlf  = lane >> 4;
  const int l16  = lane & 15;
  const size_t t0 = (size_t)blockIdx.x * 16;

  v2f ax[32];
  const float* xp = xn + (t0 + l16) * XD;
  #pragma unroll
  for (int kk = 0; kk < 32; ++kk) {
    int K0 = kk * 4 + hlf * 2;
    ax[kk].x = xp[K0];
    ax[kk].y = xp[K0 + 1];
  }

  // pass 0: gate half (cols 0..127)
  float gatev[8][8];
  #pragma unroll
  for (int nt = 0; nt < 8; ++nt) {
    v8f a;
    #pragma unroll
    for (int r = 0; r < 8; ++r) a[r] = 0.0f;
    int j = nt * 16 + l16;
    #pragma unroll
    for (int kk = 0; kk < 32; ++kk) {
      int K0 = kk * 4 + hlf * 2;
      v2f b;
      b.x = Wup[(size_t)j * XD + K0];
      b.y = Wup[(size_t)j * XD + K0 + 1];
      a = wmma4(ax[kk], b, a);
    }
    #pragma unroll
    for (int r = 0; r < 8; ++r) gatev[nt][r] = a[r];
  }

  // pass 1: upv half (cols 128..255), combine, stage act in LDS
  #pragma unroll
  for (int nt = 0; nt < 8; ++nt) {
    v8f a;
    #pragma unroll
    for (int r = 0; r < 8; ++r) a[r] = 0.0f;
    int j = 128 + nt * 16 + l16;
    #pragma unroll
    for (int kk = 0; kk < 32; ++kk) {
      int K0 = kk * 4 + hlf * 2;
      v2f b;
      b.x = Wup[(size_t)j * XD + K0];
      b.y = Wup[(size_t)j * XD + K0 + 1];
      a = wmma4(ax[kk], b, a);
    }
    #pragma unroll
    for (int r = 0; r < 8; ++r) {
      float act = gelu_exact(gatev[nt][r]) * a[r];
      lact[(r + hlf * 8) * 128 + nt * 16 + l16] = act;
    }
  }
  __syncthreads();

  // down projection + residual
  v2f aa[32];
  const float* lp = lact + l16 * 128;
  #pragma unroll
  for (int kk = 0; kk < 32; ++kk) {
    int K0 = kk * 4 + hlf * 2;
    aa[kk].x = lp[K0];
    aa[kk].y = lp[K0 + 1];
  }
  #pragma unroll
  for (int nt = 0; nt < 8; ++nt) {
    v8f a;
    #pragma unroll
    for (int r = 0; r < 8; ++r) a[r] = 0.0f;
    int j = nt * 16 + l16;
    #pragma unroll
    for (int kk = 0; kk < 32; ++kk) {
      int K0 = kk * 4 + hlf * 2;
      v2f b;
      b.x = Wdn[(size_t)j * XD + K0];
      b.y = Wdn[(size_t)j * XD + K0 + 1];
      a = wmma4(aa[kk], b, a);
    }
    #pragma unroll
    for (int r = 0; r < 8; ++r) {
      size_t row = t0 + r + hlf * 8;
      hio[row * XD + nt * 16 + l16] += a[r];
    }
  }
}

// --------------------------- vocab projection ------------------------------
// grid = T/16, block = 32. One 16x16 WMMA tile (9 valid vocab cols), K=128.
__global__ __launch_bounds__(32) void k_proj(const float* __restrict__ xn,
                                             const float* __restrict__ pw,
                                             const float* __restrict__ pb,
                                             float* __restrict__ out) {
  const int lane = threadIdx.x;
  const int hlf  = lane >> 4;
  const int l16  = lane & 15;
  const size_t t0 = (size_t)blockIdx.x * 16;

  v8f a;
  #pragma unroll
  for (int r = 0; r < 8; ++r) a[r] = 0.0f;
  const float* xp = xn + (t0 + l16) * XD;
  #pragma unroll
  for (int kk = 0; kk < 32; ++kk) {
    int K0 = kk * 4 + hlf * 2;
    v2f av;
    av.x = xp[K0];
    av.y = xp[K0 + 1];
    v2f b;
    b.x = 0.0f;
    b.y = 0.0f;
    if (l16 < 9) {
      b.x = pw[(size_t)l16 * XD + K0];
      b.y = pw[(size_t)l16 * XD + K0 + 1];
    }
    a = wmma4(av, b, a);   // full EXEC here (branch reconverged)
  }
  if (l16 < 9) {
    float bb = pb[l16];
    #pragma unroll
    for (int r = 0; r < 8; ++r) {
      size_t row = t0 + r + hlf * 8;
      out[row * 9 + l16] = a[r] + bb;
    }
  }
}

// ---------------------------------------------------------------------------
extern "C" void kernel_launch(void* const* d_in, const int* in_sizes, int n_in,
                              void* d_out, int out_size, void* d_ws, size_t ws_size,
                              hipStream_t stream) {
  (void)in_sizes; (void)n_in; (void)out_size; (void)ws_size;
  const int*   x     = (const int*)d_in[0];
  const float* emb   = (const float*)d_in[1];
  const float* ln1   = (const float*)d_in[2];
  const float* Wg    = (const float*)d_in[3];
  const float* R     = (const float*)d_in[4];
  const float* bias  = (const float*)d_in[5];
  const float* gnw   = (const float*)d_in[6];
  const float* ln2   = (const float*)d_in[7];
  const float* wup   = (const float*)d_in[8];
  const float* wdn   = (const float*)d_in[9];
  const float* postw = (const float*)d_in[10];
  const float* pw    = (const float*)d_in[11];
  const float* pb    = (const float*)d_in[12];
  float* out = (float*)d_out;

  float* h  = (float*)d_ws;                     // T x 128
  float* xb = h + (size_t)XT * XD;              // T x 128

  k_embed<<<(XT * XD + 255) / 256, 256, 0, stream>>>(x, emb, h, XT * XD);

  for (int bi = 0; bi < 2; ++bi) {
    k_ln<<<XT / 8, 256, 0, stream>>>(h, ln1 + bi * XD, xb);
    k_scan<<<dim3(XB / 16, XNH), 32, 0, stream>>>(
        xb, h,
        Wg   + (size_t)bi * 4 * XNH * XDH * XDH,
        R    + (size_t)bi * XNH * XDH * 4 * XDH,
        bias + (size_t)bi * XNH * 4 * XDH,
        gnw  + (size_t)bi * XD);
    k_ln<<<XT / 8, 256, 0, stream>>>(h, ln2 + bi * XD, xb);
    k_ffn<<<XT / 16, 32, 0, stream>>>(xb, h,
                                      wup + (size_t)bi * 2 * XD * XD,
                                      wdn + (size_t)bi * XD * XD);
  }

  k_ln<<<XT / 8, 256, 0, stream>>>(h, postw, xb);
  k_proj<<<XT / 16, 32, 0, stream>>>(xb, pw, pb, out);
}